// RotationallyDeformableConvolution_10960756539498
// MI455X (gfx1250) — compile-verified
//
#include <hip/hip_runtime.h>
#include <hip/hip_bf16.h>
#include <math.h>

typedef __attribute__((ext_vector_type(2))) float v2f;
typedef __attribute__((ext_vector_type(8))) float v8f;

#define B_   4
#define C_   256
#define H_   64
#define W_   64
#define CO_  256
#define HW_  (H_ * W_)
#define CK_  (C_ * 9)      // 2304 = GEMM K dimension

// ---------------------------------------------------------------------------
// Kernel 1: depthwise conv + channel LayerNorm + ReLU + two pointwise dots
//           -> per-pixel deformation offsets (ox, oy)
// ---------------------------------------------------------------------------
__global__ __launch_bounds__(256) void rdc_offsets_kernel(
    const float* __restrict__ x, const float* __restrict__ dwc_w,
    const float* __restrict__ ln_w, const float* __restrict__ ln_b,
    const float* __restrict__ fcd_w, const float* __restrict__ fca_w,
    float* __restrict__ ox, float* __restrict__ oy)
{
    const int pix = blockIdx.x;              // 0 .. B*H*W-1
    const int w   = pix & (W_ - 1);
    const int h   = (pix >> 6) & (H_ - 1);
    const int b   = pix >> 12;
    const int c   = threadIdx.x;

    const float* plane = x + (size_t)(b * C_ + c) * HW_;
    const float* wk    = dwc_w + c * 9;

    float hv = 0.f;
#pragma unroll
    for (int ky = 0; ky < 3; ++ky) {
        const int yy = h - 1 + ky;
        if ((unsigned)yy < (unsigned)H_) {
#pragma unroll
            for (int kx = 0; kx < 3; ++kx) {
                const int xx = w - 1 + kx;
                if ((unsigned)xx < (unsigned)W_)
                    hv = fmaf(plane[yy * W_ + xx], wk[ky * 3 + kx], hv);
            }
        }
    }

    __shared__ float s0[256];
    __shared__ float s1[256];
    s0[c] = hv;
    s1[c] = hv * hv;
    __syncthreads();
    for (int off = 128; off >= 1; off >>= 1) {
        if (c < off) { s0[c] += s0[c + off]; s1[c] += s1[c + off]; }
        __syncthreads();
    }
    const float mu  = s0[0] * (1.f / C_);
    const float var = s1[0] * (1.f / C_) - mu * mu;
    __syncthreads();                          // protect s0[0]/s1[0] before reuse

    float hn = (hv - mu) * rsqrtf(var + 1e-5f) * ln_w[c] + ln_b[c];
    hn = fmaxf(hn, 0.f);
    s0[c] = hn * fcd_w[c];
    s1[c] = hn * fca_w[c];
    __syncthreads();
    for (int off = 128; off >= 1; off >>= 1) {
        if (c < off) { s0[c] += s0[c + off]; s1[c] += s1[c + off]; }
        __syncthreads();
    }
    if (c == 0) {
        const float r  = fmaxf(s0[0], 0.f);
        const float t  = s1[0];
        const float th = t / (1.f + fabsf(t)) * 0.017453292519943295f; // pi/180
        ox[pix] = r * cosf(th);
        oy[pix] = r * sinf(th);
    }
}

// ---------------------------------------------------------------------------
// Kernel 2: fused bilinear-sampling + GEMM with V_WMMA_F32_16X16X4_F32.
// Block = 256 threads (8 waves) handles 32 pixels x all 256 output channels.
// Each wave: 2 o-tiles x 2 pixel-tiles = 4 accumulators; every A-fragment
// feeds two WMMAs. B tile is stored K-pair-interleaved in LDS so each B
// fragment is one aligned ds_load_b64 (no repack movs); row stride of 96
// floats makes both lane-halves land on disjoint LDS banks.
// ---------------------------------------------------------------------------
#define CB     16            // channels per K-chunk
#define KCH    (CB * 9)      // 144 K values per chunk
#define TP     32            // pixels per block (two 16-wide N tiles)
#define BROW   96            // LDS row stride (floats) per K-pair row

__device__ __forceinline__ float rdc_samp(const float* __restrict__ plane,
                                          int y, int xx) {
    return ((unsigned)y < (unsigned)H_ && (unsigned)xx < (unsigned)W_)
               ? plane[(y << 6) + xx] : 0.f;
}

__global__ __launch_bounds__(256) void rdc_gemm_kernel(
    const float* __restrict__ x, const float* __restrict__ dconv_w,
    const float* __restrict__ dconv_b,
    const float* __restrict__ ox, const float* __restrict__ oy,
    float* __restrict__ out)
{
    // K-pair-interleaved B tile: element (k, p) at sB[(k>>1)*BROW + p*2 + (k&1)]
    __shared__ __align__(16) float sB[(KCH / 2) * BROW];   // 72*96*4 = 27.6 KB
    __shared__ int   sIx[TP], sIy[TP];       // floor(ox), floor(oy)
    __shared__ float sFx1[TP], sFy1[TP];     // frac parts
    __shared__ float sFx0[TP], sFy0[TP];     // 1 - frac

    const int w0   = blockIdx.x * TP;
    const int h    = blockIdx.y;
    const int b    = blockIdx.z;
    const int tid  = threadIdx.x;
    const int lane = tid & 31;
    const int wave = tid >> 5;

    if (tid < TP) {
        const int   pix = (b * H_ + h) * W_ + w0 + tid;
        const float vx  = ox[pix];
        const float vy  = oy[pix];
        const float fx  = floorf(vx);
        const float fy  = floorf(vy);
        sIx[tid]  = (int)fx;       sIy[tid]  = (int)fy;
        sFx1[tid] = vx - fx;       sFy1[tid] = vy - fy;
        sFx0[tid] = 1.f - (vx - fx);
        sFy0[tid] = 1.f - (vy - fy);
    }

    v8f acc00 = {0.f,0.f,0.f,0.f,0.f,0.f,0.f,0.f};   // o-tile 0, pix-tile 0
    v8f acc01 = acc00;                               // o-tile 0, pix-tile 1
    v8f acc10 = acc00;                               // o-tile 1, pix-tile 0
    v8f acc11 = acc00;                               // o-tile 1, pix-tile 1

    const float* xb   = x + (size_t)b * C_ * HW_;
    const int    m    = lane & 15;                 // A: M index / B: N index
    const int    koff = (lane < 16) ? 0 : 2;       // K split across lane halves
    const int    o0   = wave * 32;                 // this wave's two o-tiles
    const float* wrow0 = dconv_w + (size_t)(o0 + m) * CK_;
    const float* wrow1 = wrow0 + (size_t)16 * CK_;
    // per-lane base into the interleaved B tile: row koff/2, pixel m
    const int    boff = (koff >> 1) * BROW + (m << 1);

    for (int cb = 0; cb < C_; cb += CB) {
        __syncthreads();                           // previous tile consumed

        // Cooperative bilinear gather: (CB channels x 9 taps) x 32 pixels.
        for (int e = tid; e < KCH * TP; e += 256) {
            const int ck = e >> 5;                 // 0..143 : K index in chunk
            const int p  = e & 31;                 // pixel in strip
            const int ci = ck / 9;
            const int kk = ck - ci * 9;            // tap 0..8
            const int ky = kk / 3;
            const int kx = kk - ky * 3;
            // offset interleave: dy = (k<=4 ? ox : oy), dx = (k<=3 ? ox : oy)
            const bool  uy  = (kk <= 4);
            const bool  ux  = (kk <= 3);
            const int   y0  = h - 1 + ky      + (uy ? sIx[p]  : sIy[p]);
            const int   x0i = w0 + p - 1 + kx + (ux ? sIx[p]  : sIy[p]);
            const float wy1 = uy ? sFx1[p] : sFy1[p];
            const float wy0 = uy ? sFx0[p] : sFy0[p];
            const float wx1 = ux ? sFx1[p] : sFy1[p];
            const float wx0 = ux ? sFx0[p] : sFy0[p];
            const float* plane = xb + (size_t)(cb + ci) * HW_;
            const float v00 = rdc_samp(plane, y0,     x0i);
            const float v01 = rdc_samp(plane, y0,     x0i + 1);
            const float v10 = rdc_samp(plane, y0 + 1, x0i);
            const float v11 = rdc_samp(plane, y0 + 1, x0i + 1);
            sB[(ck >> 1) * BROW + (p << 1) + (ck & 1)] =
                  wy0 * (wx0 * v00 + wx1 * v01)
                + wy1 * (wx0 * v10 + wx1 * v11);
        }
        __syncthreads();

        const float* wa = wrow0 + cb * 9;
        const float* wb = wrow1 + cb * 9;
        if (cb + CB < C_)                           // pull next weight chunk
            __builtin_prefetch(wa + KCH, 0, 3);

#pragma unroll 4
        for (int kb = 0; kb < KCH; kb += 4) {
            v2f a0; a0.x = wa[kb + koff]; a0.y = wa[kb + koff + 1];
            v2f a1; a1.x = wb[kb + koff]; a1.y = wb[kb + koff + 1];
            const float* brow = sB + (kb >> 1) * BROW + boff;
            const v2f b0 = *(const v2f*)brow;         // pixels 0..15
            const v2f b1 = *(const v2f*)(brow + 32);  // pixels 16..31
            acc00 = __builtin_amdgcn_wmma_f32_16x16x4_f32(
                        false, a0, false, b0, (short)0, acc00, false, false);
            acc01 = __builtin_amdgcn_wmma_f32_16x16x4_f32(
                        false, a0, false, b1, (short)0, acc01, false, false);
            acc10 = __builtin_amdgcn_wmma_f32_16x16x4_f32(
                        false, a1, false, b0, (short)0, acc10, false, false);
            acc11 = __builtin_amdgcn_wmma_f32_16x16x4_f32(
                        false, a1, false, b1, (short)0, acc11, false, false);
        }
    }

    // Epilogue: D layout (16x16 f32): VGPR v -> M = v + (lane<16 ? 0 : 8), N = lane&15
    const int mbase = (lane < 16) ? 0 : 8;
#pragma unroll
    for (int v = 0; v < 8; ++v) {
        const int oa = o0 + mbase + v;          // o-tile 0
        const int ob = o0 + 16 + mbase + v;     // o-tile 1
        const float ba = dconv_b[oa];
        const float bb = dconv_b[ob];
        const size_t rowa = (((size_t)b * CO_ + oa) * H_ + h) * W_ + w0;
        const size_t rowb = (((size_t)b * CO_ + ob) * H_ + h) * W_ + w0;
        out[rowa + m]      = acc00[v] + ba;
        out[rowa + 16 + m] = acc01[v] + ba;
        out[rowb + m]      = acc10[v] + bb;
        out[rowb + 16 + m] = acc11[v] + bb;
    }
}

// ---------------------------------------------------------------------------
extern "C" void kernel_launch(void* const* d_in, const int* in_sizes, int n_in,
                              void* d_out, int out_size, void* d_ws, size_t ws_size,
                              hipStream_t stream) {
    const float* x       = (const float*)d_in[0];
    const float* dwc_w   = (const float*)d_in[1];
    const float* ln_w    = (const float*)d_in[2];
    const float* ln_b    = (const float*)d_in[3];
    const float* fcd_w   = (const float*)d_in[4];
    const float* fca_w   = (const float*)d_in[5];
    const float* dconv_w = (const float*)d_in[6];
    const float* dconv_b = (const float*)d_in[7];
    float* out = (float*)d_out;

    float* ox = (float*)d_ws;                 // B*H*W floats
    float* oy = ox + (size_t)B_ * HW_;        // B*H*W floats (128 KB total)

    rdc_offsets_kernel<<<B_ * HW_, 256, 0, stream>>>(
        x, dwc_w, ln_w, ln_b, fcd_w, fca_w, ox, oy);

    dim3 grid(W_ / TP, H_, B_);               // 2 x 64 x 4 = 512 blocks
    rdc_gemm_kernel<<<grid, 256, 0, stream>>>(
        x, dconv_w, dconv_b, ox, oy, out);
}